// RagContrastiveWeights_56882546868664
// MI455X (gfx1250) — compile-verified
//
#include <hip/hip_runtime.h>
#include <hip/hip_bf16.h>

// ---------------------------------------------------------------------------
// RAG contrastive loss for MI455X (gfx1250, wave32, WMMA).
//
// Shapes (fixed by the reference):
//   embeddings [B=4][E=32][H=512][W=512] f32
//   sp_seg     [B][1][H][W]              i32  (values in [0, C=2048))
//   edges      [B][2][NE=8192]           i32
//   weights    [B][NE]                   f32
//   out        scalar f32
//
// Strategy:
//  K1: per-block segment histogram in LDS (C * 33 floats = 270336 B, uses the
//      CDNA5 320KB WGP LDS; stride 33 => conflict-free ds_add_f32 banks),
//      merged to global sums/counts with atomics.
//  K2: means = sums / max(cnt,1), L2-normalized; also writes 1/max(cnt,1)
//      so the hot intra pass multiplies instead of dividing.
//  K3: intra term: 16 pixel-dot-products per wave batched through
//      V_WMMA_F32_16X16X4_F32 (8 K-chunks), branchless diagonal extraction.
//  K4: inter term: 16 edge-dot-products per wave, same WMMA-diagonal trick.
// ---------------------------------------------------------------------------

#define B_  4
#define E_  32
#define H_  512
#define W_  512
#define N_  (H_ * W_)        // 262144 pixels per image
#define C_  2048             // superpixels
#define NE_ 8192             // edges per image

#define DELTA_VAR  0.1f
#define DELTA_DIST 0.3f
#define ALPHA_     1.0f
#define BETA_      1.0f

#define LDS_STRIDE   33                         // 32 channel sums + 1 count, odd => conflict free
#define LDS_FLOATS   (C_ * LDS_STRIDE)          // 67584 floats
#define LDS_BYTES    (LDS_FLOATS * 4)           // 270336 bytes <= 320KB WGP LDS
#define BLOCKS_PER_IMG 32
#define PIX_PER_BLOCK  (N_ / BLOCKS_PER_IMG)    // 8192

typedef float v2f __attribute__((ext_vector_type(2)));
typedef float v8f __attribute__((ext_vector_type(8)));

__device__ __forceinline__ v8f wmma_f32_4(v2f a, v2f b, v8f c) {
  // D = A(16x4 f32) * B(4x16 f32) + C ; emits v_wmma_f32_16x16x4_f32
  return __builtin_amdgcn_wmma_f32_16x16x4_f32(false, a, false, b, (short)0, c,
                                               false, false);
}

// Extract component `comp` (0..7) from a v8f without scratch spills.
__device__ __forceinline__ float v8f_get(v8f v, int comp) {
  float r = 0.f;
#pragma unroll
  for (int j = 0; j < 8; ++j) r = (comp == j) ? v[j] : r;
  return r;
}

__device__ __forceinline__ float wave32_sum(float v) {
#pragma unroll
  for (int off = 16; off >= 1; off >>= 1) v += __shfl_xor(v, off, 32);
  return v;
}

// ---------------------------------------------------------------------------
// K1: segment sums + counts, LDS-resident per-block histogram.
// grid: B_ * BLOCKS_PER_IMG blocks, 1024 threads, LDS_BYTES dynamic LDS.
// ---------------------------------------------------------------------------
__global__ __launch_bounds__(1024) void seg_accum_kernel(
    const float* __restrict__ emb, const int* __restrict__ seg,
    float* __restrict__ counts, float* __restrict__ sums) {
  extern __shared__ float lds[];
  const int tid = threadIdx.x;
  const int b   = blockIdx.x / BLOCKS_PER_IMG;
  const int blk = blockIdx.x % BLOCKS_PER_IMG;

  for (int i = tid; i < LDS_FLOATS; i += 1024) lds[i] = 0.f;
  __syncthreads();

  const float* embB = emb + (size_t)b * E_ * N_;
  const int*   segB = seg + (size_t)b * N_;
  const int p0 = blk * PIX_PER_BLOCK;

  for (int p = p0 + tid; p < p0 + PIX_PER_BLOCK; p += 1024) {
    const int c = segB[p];
    float* row = &lds[c * LDS_STRIDE];
#pragma unroll
    for (int e = 0; e < E_; ++e) {
      atomicAdd(&row[e], embB[(size_t)e * N_ + p]);   // ds_add_f32
    }
    atomicAdd(&row[E_], 1.0f);                        // count slot
  }
  __syncthreads();

  float* cntB = counts + (size_t)b * C_;
  float* sumB = sums + (size_t)b * C_ * E_;
  for (int i = tid; i < LDS_FLOATS; i += 1024) {
    const float v = lds[i];
    if (v != 0.f) {
      const int c = i / LDS_STRIDE, j = i % LDS_STRIDE;
      if (j < E_) atomicAdd(&sumB[c * E_ + j], v);
      else        atomicAdd(&cntB[c], v);
    }
  }
}

// ---------------------------------------------------------------------------
// K2: means = normalize(sums / max(cnt,1)); inv_counts = 1/max(cnt,1).
// One thread per (b,c).
// ---------------------------------------------------------------------------
__global__ __launch_bounds__(256) void means_kernel(
    const float* __restrict__ counts, const float* __restrict__ sums,
    float* __restrict__ means, float* __restrict__ inv_counts) {
  const int t = blockIdx.x * 256 + threadIdx.x;
  if (t >= B_ * C_) return;
  const float inv_c = 1.0f / fmaxf(counts[t], 1.0f);
  inv_counts[t] = inv_c;
  const float* s = sums + (size_t)t * E_;
  float m[E_], nrm2 = 0.f;
#pragma unroll
  for (int e = 0; e < E_; ++e) { m[e] = s[e] * inv_c; nrm2 += m[e] * m[e]; }
  const float inv_n = 1.0f / fmaxf(sqrtf(nrm2), 1e-10f);
  float* d = means + (size_t)t * E_;
#pragma unroll
  for (int e = 0; e < E_; ++e) d[e] = m[e] * inv_n;
}

// ---------------------------------------------------------------------------
// K3: intra term.  Each wave handles groups of 16 pixels; per group, 8x
// v_wmma_f32_16x16x4_f32 accumulate D = Emb(16x32) * MeansSel(32x16);
// diagonal n gives <emb[p_n], means[seg[p_n]]>.  Branchless extraction.
// grid: B_*64 blocks of 256 threads (8 waves); 512 waves/image, 16384 groups.
// ---------------------------------------------------------------------------
__global__ __launch_bounds__(256) void intra_kernel(
    const float* __restrict__ emb, const int* __restrict__ seg,
    const float* __restrict__ inv_counts, const float* __restrict__ means,
    float* __restrict__ out) {
  const int lane = threadIdx.x & 31;
  const int wave = threadIdx.x >> 5;                 // 0..7
  const int b    = blockIdx.x >> 6;                  // /64
  const int blkI = blockIdx.x & 63;
  const int waveInImg = blkI * 8 + wave;             // 0..511

  const int m = lane & 15;     // pixel-in-group (A row / B col)
  const int h = lane >> 4;     // K half-select per ISA A/B layout

  const float* embB  = emb        + (size_t)b * E_ * N_;
  const int*   segB  = seg        + (size_t)b * N_;
  const float* icntB = inv_counts + (size_t)b * C_;
  const float* meanB = means      + (size_t)b * C_ * E_;

  const bool  diag = (lane < 8) || (lane >= 24);
  const int   comp = (lane < 8) ? lane : (lane - 24);
  const float dmask = diag ? 1.0f : 0.0f;

  float partial = 0.f;
  for (int g = waveInImg; g < N_ / 16; g += 512) {   // uniform per wave
    const int p = g * 16 + m;
    const int s = segB[p];
    const float invc = icntB[s];                     // per-pixel constant
    const float* mr = meanB + (size_t)s * E_;
    const float* er = embB + p;

    v8f acc = {};
#pragma unroll
    for (int kk = 0; kk < 8; ++kk) {
      const int ch = kk * 4 + 2 * h;
      v2f a = { er[(size_t)ch * N_], er[(size_t)(ch + 1) * N_] };
      v2f bb = { mr[ch], mr[ch + 1] };
      acc = wmma_f32_4(a, bb, acc);
    }
    const float dot = v8f_get(acc, comp);            // junk on non-diag lanes
    const float d = 1.0f - dot;
    partial += dmask * fmaxf(d - DELTA_VAR, 0.0f) * invc;
  }
  partial = wave32_sum(partial);
  if (lane == 0) atomicAdd(out, partial * (BETA_ / (float)C_));
}

// ---------------------------------------------------------------------------
// K4: inter (edge) term.  16 edges per wave-group via WMMA diagonal of
// MeansA(16x32) * MeansB(32x16).  grid: B_ blocks of 256 threads.
// ---------------------------------------------------------------------------
__global__ __launch_bounds__(256) void inter_kernel(
    const int* __restrict__ edges, const float* __restrict__ weights,
    const float* __restrict__ means, float* __restrict__ out) {
  const int lane = threadIdx.x & 31;
  const int wave = threadIdx.x >> 5;                 // 0..7
  const int b    = blockIdx.x;

  const int n = lane & 15;     // edge-in-group
  const int h = lane >> 4;

  const int*   e0B = edges + (size_t)b * 2 * NE_;
  const int*   e1B = e0B + NE_;
  const float* wB  = weights + (size_t)b * NE_;
  const float* meanB = means + (size_t)b * C_ * E_;

  const bool  diag = (lane < 8) || (lane >= 24);
  const int   comp = (lane < 8) ? lane : (lane - 24);
  const float dmask = diag ? 1.0f : 0.0f;

  float partial = 0.f;
  for (int g = wave; g < NE_ / 16; g += 8) {         // 64 groups per wave
    const int e = g * 16 + n;
    const float w = wB[e];
    const float* r0 = meanB + (size_t)e0B[e] * E_;   // A row m = n
    const float* r1 = meanB + (size_t)e1B[e] * E_;   // B col n

    v8f acc = {};
#pragma unroll
    for (int kk = 0; kk < 8; ++kk) {
      const int ch = kk * 4 + 2 * h;
      v2f a = { r0[ch], r0[ch + 1] };
      v2f bb = { r1[ch], r1[ch + 1] };
      acc = wmma_f32_4(a, bb, acc);
    }
    const float dot = v8f_get(acc, comp);
    const float inter = 1.0f - dot;
    partial += dmask * fmaxf(DELTA_DIST - inter * w, 0.0f);
  }
  partial = wave32_sum(partial);
  if (lane == 0) atomicAdd(out, partial * (ALPHA_ / (float)NE_));
}

// ---------------------------------------------------------------------------
// Host launcher.  ws layout (floats):
//   [0, B*C)                                 counts
//   [B*C, B*C + B*C*E)                       sums
//   [B*C + B*C*E, B*C + 2*B*C*E)             means
//   [B*C + 2*B*C*E, 2*B*C + 2*B*C*E)         inv_counts
// total ~2.06 MB.
// ---------------------------------------------------------------------------
extern "C" void kernel_launch(void* const* d_in, const int* in_sizes, int n_in,
                              void* d_out, int out_size, void* d_ws, size_t ws_size,
                              hipStream_t stream) {
  const float* emb     = (const float*)d_in[0];
  const int*   seg     = (const int*)d_in[1];
  const int*   edges   = (const int*)d_in[2];
  const float* weights = (const float*)d_in[3];
  float* out = (float*)d_out;

  float* counts  = (float*)d_ws;
  float* sums    = counts + (size_t)B_ * C_;
  float* means   = sums + (size_t)B_ * C_ * E_;
  float* invcnt  = means + (size_t)B_ * C_ * E_;

  // Allow 270 KB dynamic LDS (CDNA5 WGP has 320 KB). Not a stream op; safe
  // under graph capture and idempotent.
  (void)hipFuncSetAttribute((const void*)seg_accum_kernel,
                            hipFuncAttributeMaxDynamicSharedMemorySize,
                            LDS_BYTES);

  // Zero counts+sums and the output accumulator.
  hipMemsetAsync(d_ws, 0, (size_t)(B_ * C_ + B_ * C_ * E_) * sizeof(float), stream);
  hipMemsetAsync(d_out, 0, sizeof(float), stream);

  seg_accum_kernel<<<dim3(B_ * BLOCKS_PER_IMG), dim3(1024), LDS_BYTES, stream>>>(
      emb, seg, counts, sums);

  means_kernel<<<dim3((B_ * C_ + 255) / 256), dim3(256), 0, stream>>>(
      counts, sums, means, invcnt);

  intra_kernel<<<dim3(B_ * 64), dim3(256), 0, stream>>>(
      emb, seg, invcnt, means, out);

  inter_kernel<<<dim3(B_), dim3(256), 0, stream>>>(
      edges, weights, means, out);
}